// EdgeSAGE_16509854286680
// MI455X (gfx1250) — compile-verified
//
#include <hip/hip_runtime.h>
#include <math.h>

typedef __attribute__((ext_vector_type(2))) float v2f;
typedef __attribute__((ext_vector_type(8))) float v8f;

#define N_NODES 100000
#define HDIM 16

// ---------------------------------------------------------------------------
// Zero-fill workspace regions (agg buffers + counts) each call.
// ---------------------------------------------------------------------------
__global__ void zero_f32(float* __restrict__ p, long long n) {
    long long t = (long long)blockIdx.x * blockDim.x + threadIdx.x;
    long long stride = (long long)gridDim.x * blockDim.x;
    for (; t < n; t += stride) p[t] = 0.0f;
}

// ---------------------------------------------------------------------------
// Node feature projection with V_WMMA_F32_16X16X4_F32.
// One wave32 handles a tile of 16 nodes. Computes BOTH
//   outl = X @ Wl            (16 x K  @  K x 16)
//   outr = X @ Wr + bias
// A fragment (16x4 f32, 2 VGPRs): lane m = lane&15 is row M; lanes 0-15 hold
// K = k0,k0+1; lanes 16-31 hold K = k0+2,k0+3 (per CDNA5 ISA 7.12.2).
// B fragment mirrors the same (lane, k) striping from row-major W[K][16].
// D fragment (16x16 f32, 8 VGPRs): vgpr v holds row M = v + 8*(lane>=16),
// column N = lane&15.
// ---------------------------------------------------------------------------
template <int K>
__global__ __launch_bounds__(256) void node_transform(
    const float* __restrict__ xin, const float* __restrict__ Wl,
    const float* __restrict__ Wr,  const float* __restrict__ bias,
    float* __restrict__ outl, float* __restrict__ outr, int nNodes)
{
    const int wave  = blockIdx.x * (blockDim.x >> 5) + (threadIdx.x >> 5);
    const int lane  = threadIdx.x & 31;
    const int tile0 = wave * 16;
    if (tile0 >= nNodes) return;          // wave-uniform exit: EXEC stays full

    const int m  = lane & 15;             // A row / B column / D column
    const int kh = (lane >> 4) << 1;      // 0 for lanes 0-15, 2 for lanes 16-31

    int rowA = tile0 + m;
    if (rowA >= nNodes) rowA = nNodes - 1;   // clamp partial tile (dup loads ok)
    const float* xrow = xin + (long long)rowA * K;

    v8f accl = {0.f,0.f,0.f,0.f,0.f,0.f,0.f,0.f};
    v8f accr = {0.f,0.f,0.f,0.f,0.f,0.f,0.f,0.f};

#pragma unroll
    for (int k0 = 0; k0 < K; k0 += 4) {
        const int kr = k0 + kh;
        v2f a = *(const v2f*)(xrow + kr);                       // x[m][kr..kr+1]
        v2f bl = { Wl[kr * HDIM + m], Wl[(kr + 1) * HDIM + m] };
        v2f br = { Wr[kr * HDIM + m], Wr[(kr + 1) * HDIM + m] };
        accl = __builtin_amdgcn_wmma_f32_16x16x4_f32(
                   false, a, false, bl, (short)0, accl, false, false);
        accr = __builtin_amdgcn_wmma_f32_16x16x4_f32(
                   false, a, false, br, (short)0, accr, false, false);
    }

    const float bv    = bias[m];          // bias along N (output-feature) dim
    const int   mbase = (lane >> 4) * 8;  // D-matrix row base per lane half
#pragma unroll
    for (int v = 0; v < 8; ++v) {
        const int row = tile0 + mbase + v;
        if (row < nNodes) {
            outl[row * HDIM + m] = accl[v];
            outr[row * HDIM + m] = accr[v] + bv;
        }
    }
}

// ---------------------------------------------------------------------------
// Edge aggregation: agg[dst] += y[src] (16 features/edge, one thread per
// edge-feature). fp32 global atomics resolve in L2 (agg buffer = 6.4 MB).
// Optionally accumulates in-degree counts (layer 1 only).
// ---------------------------------------------------------------------------
__global__ __launch_bounds__(256) void edge_aggregate(
    const float* __restrict__ y, const int* __restrict__ eidx,
    float* __restrict__ agg, float* __restrict__ cnt, int E, int withCount)
{
    const long long total  = (long long)E * HDIM;
    const long long stride = (long long)gridDim.x * blockDim.x;
    for (long long t = (long long)blockIdx.x * blockDim.x + threadIdx.x;
         t < total; t += stride) {
        const int e = (int)(t >> 4);
        const int j = (int)(t & 15);
        const int s = eidx[e];        // src row of edge_index
        const int d = eidx[E + e];    // dst row of edge_index
        atomicAdd(agg + (long long)d * HDIM + j, y[(long long)s * HDIM + j]);
        if (withCount && j == 0) atomicAdd(cnt + d, 1.0f);
    }
}

// ---------------------------------------------------------------------------
// Combine: out = [relu]( agg / max(cnt,1) + yr )
// ---------------------------------------------------------------------------
__global__ __launch_bounds__(256) void combine(
    const float* __restrict__ agg, const float* __restrict__ yr,
    const float* __restrict__ cnt, float* __restrict__ out,
    int nNodes, int doRelu)
{
    const int t = blockIdx.x * blockDim.x + threadIdx.x;
    if (t >= nNodes * HDIM) return;
    const int node = t >> 4;
    float c = cnt[node];
    c = c > 1.0f ? c : 1.0f;
    float v = agg[t] / c + yr[t];
    if (doRelu) v = v > 0.0f ? v : 0.0f;
    out[t] = v;
}

// ---------------------------------------------------------------------------
// Decode: sigmoid( dot(h2[a], h2[b]) ), 16-dim dot with float4 loads.
// ---------------------------------------------------------------------------
__global__ __launch_bounds__(256) void decode_edges(
    const float* __restrict__ h, const int* __restrict__ didx,
    float* __restrict__ out, int P)
{
    const int t = blockIdx.x * blockDim.x + threadIdx.x;
    if (t >= P) return;
    const int a = didx[t];
    const int b = didx[P + t];
    const float4* ha = (const float4*)(h + (long long)a * HDIM);
    const float4* hb = (const float4*)(h + (long long)b * HDIM);
    float s = 0.0f;
#pragma unroll
    for (int i = 0; i < 4; ++i) {
        float4 u = ha[i];
        float4 w = hb[i];
        s += u.x * w.x + u.y * w.y + u.z * w.z + u.w * w.w;
    }
    out[t] = 1.0f / (1.0f + __expf(-s));
}

// ---------------------------------------------------------------------------
// Host-side launcher
// ---------------------------------------------------------------------------
extern "C" void kernel_launch(void* const* d_in, const int* in_sizes, int n_in,
                              void* d_out, int out_size, void* d_ws, size_t ws_size,
                              hipStream_t stream) {
    const float* x   = (const float*)d_in[0];   // [N,128]
    const float* W1l = (const float*)d_in[1];   // [128,16]
    const float* b1  = (const float*)d_in[2];   // [16]
    const float* W1r = (const float*)d_in[3];   // [128,16]
    const float* W2l = (const float*)d_in[4];   // [16,16]
    const float* b2  = (const float*)d_in[5];   // [16]
    const float* W2r = (const float*)d_in[6];   // [16,16]
    const int* eidx  = (const int*)d_in[7];     // [2,E]
    const int* didx  = (const int*)d_in[8];     // [2,P]
    float* out = (float*)d_out;

    const int N = in_sizes[0] / 128;            // 100000
    const int E = in_sizes[7] / 2;              // 1.6M
    const int P = in_sizes[8] / 2;              // 2.0M
    const long long NH = (long long)N * HDIM;

    // Workspace layout (reused across phases):
    //   buf0: y1l -> h1 -> h2
    //   buf1: y1r -> z_l
    //   buf2: agg1 -> z_r
    //   buf3: agg2
    //   cnt : in-degree counts
    float* buf0 = (float*)d_ws;
    float* buf1 = buf0 + NH;
    float* buf2 = buf1 + NH;
    float* buf3 = buf2 + NH;
    float* cnt  = buf3 + NH;

    const int TB = 256;
    const int wavesPerBlock = TB / 32;
    const int tiles = (N + 15) / 16;
    const int xfBlocks = (tiles + wavesPerBlock - 1) / wavesPerBlock;
    const long long edgeWork = (long long)E * HDIM;
    const int edgeBlocks = (int)((edgeWork + TB - 1) / TB);
    const int nodeBlocks = (int)((NH + TB - 1) / TB);
    const int decBlocks  = (P + TB - 1) / TB;

    // 1) zero agg1, agg2, cnt (contiguous: buf2 .. cnt end)
    zero_f32<<<2048, TB, 0, stream>>>(buf2, 2 * NH + N);

    // 2) layer-1 projection (WMMA): y1l = x@W1l ; y1r = x@W1r + b1
    node_transform<128><<<xfBlocks, TB, 0, stream>>>(x, W1l, W1r, b1,
                                                     buf0, buf1, N);

    // 3) agg1[d] += y1l[s], plus degree counts
    edge_aggregate<<<edgeBlocks, TB, 0, stream>>>(buf0, eidx, buf2, cnt, E, 1);

    // 4) h1 = relu(agg1/cnt + y1r)  -> buf0
    combine<<<nodeBlocks, TB, 0, stream>>>(buf2, buf1, cnt, buf0, N, 1);

    // 5) layer-2 projection (WMMA): z_l = h1@W2l -> buf1 ; z_r = h1@W2r+b2 -> buf2
    node_transform<16><<<xfBlocks, TB, 0, stream>>>(buf0, W2l, W2r, b2,
                                                    buf1, buf2, N);

    // 6) agg2[d] += z_l[s]
    edge_aggregate<<<edgeBlocks, TB, 0, stream>>>(buf1, eidx, buf3, cnt, E, 0);

    // 7) h2 = agg2/cnt + z_r  -> buf0
    combine<<<nodeBlocks, TB, 0, stream>>>(buf3, buf2, cnt, buf0, N, 0);

    // 8) logits -> sigmoid
    decode_edges<<<decBlocks, TB, 0, stream>>>(buf0, didx, out, P);
}